// HSVAdversary_1202590843526
// MI455X (gfx1250) — compile-verified
//
#include <hip/hip_runtime.h>
#include <stdint.h>

// ---------------------------------------------------------------------------
// Fused GaussianBlur(7x7, sigma=2, reflect) + HSV-perturbation adversary.
// Memory-bound (a few FLOP/byte): floor = 604 MB / 23.3 TB/s ~= 26 us.
// Strategy: single fused pass; stage vars halo tiles to LDS via CDNA5
// async global->LDS loads (ASYNCcnt path); non-temporal streams for
// images/out; all HSV math replicates JAX semantics exactly in fp32.
// ---------------------------------------------------------------------------

#define IMG_H   512
#define IMG_W   512
#define CH      3
#define TILE_W  64
#define TILE_H  32
#define HALO    3
#define RW      (TILE_W + 2 * HALO)   // 70
#define RH      (TILE_H + 2 * HALO)   // 38
#define NTHREADS 256

// Gaussian weights, ks=7 sigma=2 (normalized, matches jnp in fp32 to ~1 ulp)
#define KW0 0.07015933f
#define KW1 0.13107488f
#define KW2 0.19071282f
#define KW3 0.21610594f

typedef __attribute__((address_space(1))) int* global_i32_ptr;
typedef __attribute__((address_space(3))) int* lds_i32_ptr;

__device__ __forceinline__ int reflect_idx(int i, int n) {
    i = (i < 0) ? -i : i;
    i = (i >= n) ? (2 * n - 2 - i) : i;
    return i;
}

// CDNA5 async global->LDS copy of one dword (tracked by ASYNCcnt).
__device__ __forceinline__ void async_load_f32(const float* gp, float* lp) {
#if __has_builtin(__builtin_amdgcn_global_load_async_to_lds_b32)
    __builtin_amdgcn_global_load_async_to_lds_b32(
        (global_i32_ptr)(uintptr_t)gp,
        (lds_i32_ptr)(uint32_t)(uintptr_t)lp,   // low 32 bits = LDS offset
        0, 0);
#else
    uint32_t lds_off = (uint32_t)(uintptr_t)lp;
    uint64_t gaddr   = (uint64_t)(uintptr_t)gp;
    asm volatile("global_load_async_to_lds_b32 %0, %1, off"
                 :: "v"(lds_off), "v"(gaddr) : "memory");
#endif
}

__device__ __forceinline__ void wait_async_all() {
#if __has_builtin(__builtin_amdgcn_s_wait_asynccnt)
    __builtin_amdgcn_s_wait_asynccnt(0);
#else
    asm volatile("s_wait_asynccnt 0x0" ::: "memory");
#endif
}

extern "C" __global__ __launch_bounds__(NTHREADS)
void hsv_adversary_fused(const float* __restrict__ vars,
                         const float* __restrict__ images,
                         float* __restrict__ out)
{
    __shared__ float s_raw[CH][RH][RW];       // reflect-padded vars tile
    __shared__ float s_hb [CH][RH][TILE_W];   // horizontally blurred

    const int tid = threadIdx.x;
    const int tx0 = blockIdx.x * TILE_W;
    const int ty0 = blockIdx.y * TILE_H;
    const int b   = blockIdx.z;

    const float Kf[7] = {KW0, KW1, KW2, KW3, KW2, KW1, KW0};

    const size_t plane = (size_t)IMG_H * IMG_W;
    const float* vbase = vars + (size_t)b * CH * plane;

    // ---- Phase 1: async-stage reflect-padded vars halo tile into LDS ----
    const int nwords = CH * RH * RW;          // 7980
    for (int idx = tid; idx < nwords; idx += NTHREADS) {
        int c   = idx / (RH * RW);
        int r   = idx - c * (RH * RW);
        int row = r / RW;
        int col = r - row * RW;
        int gy  = reflect_idx(ty0 + row - HALO, IMG_H);
        int gx  = reflect_idx(tx0 + col - HALO, IMG_W);
        async_load_f32(vbase + c * plane + (size_t)gy * IMG_W + gx,
                       &s_raw[c][row][col]);
    }
    wait_async_all();
    __syncthreads();

    // ---- Phase 2: horizontal 7-tap ----
    const int nh = CH * RH * TILE_W;          // 7296
    for (int idx = tid; idx < nh; idx += NTHREADS) {
        int c   = idx / (RH * TILE_W);
        int r   = idx - c * (RH * TILE_W);
        int row = r >> 6;                     // / TILE_W
        int x   = r & (TILE_W - 1);
        float acc = 0.f;
        #pragma unroll
        for (int t = 0; t < 7; ++t)
            acc = fmaf(Kf[t], s_raw[c][row][x + t], acc);
        s_hb[c][row][x] = acc;
    }
    __syncthreads();

    // ---- Phase 3: vertical 7-tap + HSV adversary ----
    const float* ibase = images + (size_t)b * CH * plane;
    float*       obase = out    + (size_t)b * CH * plane;

    for (int p = tid; p < TILE_W * TILE_H; p += NTHREADS) {
        int x = p & (TILE_W - 1);
        int y = p >> 6;

        float dv[3];
        #pragma unroll
        for (int c = 0; c < 3; ++c) {
            float acc = 0.f;
            #pragma unroll
            for (int t = 0; t < 7; ++t)
                acc = fmaf(Kf[t], s_hb[c][y + t][x], acc);
            dv[c] = acc;
        }

        size_t off = (size_t)(ty0 + y) * IMG_W + (tx0 + x);
        float r  = __builtin_nontemporal_load(ibase + off);
        float g  = __builtin_nontemporal_load(ibase + plane + off);
        float bb = __builtin_nontemporal_load(ibase + 2 * plane + off);

        // rgb2hsv (replicates jnp.argmax first-max semantics + python %6)
        float cmax  = fmaxf(r, fmaxf(g, bb));
        float cmin  = fminf(r, fminf(g, bb));
        float delta = cmax - cmin;
        float ds    = (delta == 0.f) ? 1.f : delta;
        float h;
        if (r >= g && r >= bb) {              // argmax == 0
            h = (g - bb) / ds;
            if (h < 0.f) h += 6.f;            // python mod 6 (|h|<=1)
        } else if (g > r && g >= bb) {        // argmax == 1
            h = (bb - r) / ds + 2.f;
        } else {                              // argmax == 2
            h = (r - g) / ds + 4.f;
        }
        h = (delta == 0.f) ? 0.f : h;
        h *= (1.f / 6.f);
        float s = (cmax == 0.f) ? 0.f : delta / cmax;
        float v = cmax;

        // perturb + clip
        h = fminf(fmaxf(h + dv[0], 0.f), 1.f);
        s = fminf(fmaxf(s + dv[1], 0.f), 1.f);
        v = fminf(fmaxf(v + dv[2], 0.f), 1.f);

        // hsv2rgb
        float c6 = h * 6.f;
        float cc = v * s;
        float xx = cc * (1.f - fabsf(fmodf(c6, 2.f) - 1.f));
        float m  = v - cc;
        int i6 = ((int)floorf(c6)) % 6;       // c6 in [0,6]
        float r2, g2, b2;
        switch (i6) {
            case 0:  r2 = cc;  g2 = xx;  b2 = 0.f; break;
            case 1:  r2 = xx;  g2 = cc;  b2 = 0.f; break;
            case 2:  r2 = 0.f; g2 = cc;  b2 = xx;  break;
            case 3:  r2 = 0.f; g2 = xx;  b2 = cc;  break;
            case 4:  r2 = xx;  g2 = 0.f; b2 = cc;  break;
            default: r2 = cc;  g2 = 0.f; b2 = xx;  break;
        }
        r2 = fminf(fmaxf(r2 + m, 0.f), 1.f);
        g2 = fminf(fmaxf(g2 + m, 0.f), 1.f);
        b2 = fminf(fmaxf(b2 + m, 0.f), 1.f);

        __builtin_nontemporal_store(r2, obase + off);
        __builtin_nontemporal_store(g2, obase + plane + off);
        __builtin_nontemporal_store(b2, obase + 2 * plane + off);
    }
}

extern "C" void kernel_launch(void* const* d_in, const int* in_sizes, int n_in,
                              void* d_out, int out_size, void* d_ws, size_t ws_size,
                              hipStream_t stream) {
    const float* vars   = (const float*)d_in[0];
    const float* images = (const float*)d_in[1];
    float*       out    = (float*)d_out;
    (void)d_ws; (void)ws_size; (void)n_in; (void)out_size;

    int B = in_sizes[0] / (CH * IMG_H * IMG_W);   // 64
    dim3 grid(IMG_W / TILE_W, IMG_H / TILE_H, B); // (8, 16, 64)
    hsv_adversary_fused<<<grid, NTHREADS, 0, stream>>>(vars, images, out);
}